// IrregularGrid_21526376087721
// MI455X (gfx1250) — compile-verified
//
#include <hip/hip_runtime.h>
#include <hip/hip_bf16.h>
#include <math.h>

// Plenoxels-style volume renderer for MI455X (gfx1250, wave32).
// One block per ray. Lane-per-sample gather + trilinear blend into feat[28],
// then per-16-sample-tile SH contraction via chained V_WMMA_F32_16X16X4_F32
// (B = per-ray SH-basis selection matrix), then per-ray transmittance scan.

typedef float v2f __attribute__((ext_vector_type(2)));
typedef float v8f __attribute__((ext_vector_type(8)));

namespace {
constexpr int   kRes = 192;
constexpr float kVL  = 2.0f / 191.0f;   // VOXEL_LEN
constexpr float kHVL = 1.0f / 191.0f;   // VOXEL_LEN / 2
constexpr int   kNT  = 128;             // threads per block (4 waves)
constexpr int   kFP  = 29;              // feat LDS row stride (coprime with 32 banks)

constexpr float kC0  = 0.28209479177387814f;
constexpr float kC1  = 0.4886025119029199f;
constexpr float kC20 = 1.0925484305920792f;
constexpr float kC21 = -1.0925484305920792f;
constexpr float kC22 = 0.31539156525252005f;
constexpr float kC23 = -1.0925484305920792f;
constexpr float kC24 = 0.5462742152960396f;
} // namespace

// B-matrix element for the SH contraction: Bmat[g][n].
// g = global feature channel (0..27), n = output column (0..15).
// Columns 0..2 = R,G,B (dot with basis), column 3 = sigma passthrough.
// g is a compile-time constant at every call site -> folds to a select on n.
__device__ __forceinline__ float sh_bsel(int g, int n,
    float b0, float b1, float b2, float b3, float b4,
    float b5, float b6, float b7, float b8)
{
    if (g == 27) return (n == 3) ? 1.0f : 0.0f;
    if (g > 27)  return 0.0f;
    const int c = g / 9;
    const int k = g - 9 * c;
    float v = b0;
    if (k == 1) v = b1; else if (k == 2) v = b2; else if (k == 3) v = b3;
    else if (k == 4) v = b4; else if (k == 5) v = b5; else if (k == 6) v = b6;
    else if (k == 7) v = b7; else if (k == 8) v = b8;
    return (c == n) ? v : 0.0f;
}

__global__ __launch_bounds__(kNT) void plenoxel_render_kernel(
    const float* __restrict__ rays_o,
    const float* __restrict__ rays_d,
    const float* __restrict__ grid_data,   // [RES^3, 28]
    const int*   __restrict__ grid_idx,    // [RES,RES,RES]
    float* __restrict__ out,               // rgb_map[B,3] | alpha[B,ns-1] | depth[B]
    int B, int ns)
{
    extern __shared__ float smem[];
    float* s_sig   = smem;               // [ns]
    float* s_intr  = smem + ns;          // [ns]
    float* s_alpha = smem + 2 * ns;      // [ns]
    float* s_rgb   = smem + 3 * ns;      // [ns][3]
    float* s_feat  = smem + 6 * ns;      // [4 waves][32][kFP]

    const int b    = blockIdx.x;
    const int tid  = threadIdx.x;
    const int wave = tid >> 5;
    const int lane = tid & 31;

    const float ox = rays_o[3 * b + 0], oy = rays_o[3 * b + 1], oz = rays_o[3 * b + 2];
    const float dx = rays_d[3 * b + 0], dy = rays_d[3 * b + 1], dz = rays_d[3 * b + 2];

    // --- AABB slab intersection (matches reference) ---
    const float opx = (1.0f - ox) / dx, onx = (-1.0f - ox) / dx;
    const float opy = (1.0f - oy) / dy, ony = (-1.0f - oy) / dy;
    const float opz = (1.0f - oz) / dz, onz = (-1.0f - oz) / dz;
    const float t_in  = fmaxf(fmaxf(fminf(opx, onx), fminf(opy, ony)), fminf(opz, onz));
    const float t_out = fminf(fminf(fmaxf(opx, onx), fmaxf(opy, ony)), fmaxf(opz, onz));

    const float dn  = sqrtf(dx * dx + dy * dy + dz * dz);
    const float inv = 1.0f / dn;
    const float ux = dx * inv, uy = dy * inv, uz = dz * inv;

    // --- degree-2 SH basis for this ray ---
    const float B0 = kC0;
    const float B1 = -kC1 * uy;
    const float B2 =  kC1 * uz;
    const float B3 = -kC1 * ux;
    const float B4 = kC20 * ux * uy;
    const float B5 = kC21 * uy * uz;
    const float B6 = kC22 * (2.0f * uz * uz - ux * ux - uy * uy);
    const float B7 = kC23 * ux * uz;
    const float B8 = kC24 * (ux * ux - uy * uy);

    // --- per-ray WMMA B-matrix fragments (constant across the sample loop) ---
    // B 4x16 layout: N = lane&15; VGPR0 holds K=(lane<16?0:2), VGPR1 holds K+1.
    float bw0[7], bw1[7];
    {
        const int n = lane & 15;
        if (lane < 16) {
#pragma unroll
            for (int j = 0; j < 7; ++j) {
                bw0[j] = sh_bsel(4 * j + 0, n, B0, B1, B2, B3, B4, B5, B6, B7, B8);
                bw1[j] = sh_bsel(4 * j + 1, n, B0, B1, B2, B3, B4, B5, B6, B7, B8);
            }
        } else {
#pragma unroll
            for (int j = 0; j < 7; ++j) {
                bw0[j] = sh_bsel(4 * j + 2, n, B0, B1, B2, B3, B4, B5, B6, B7, B8);
                bw1[j] = sh_bsel(4 * j + 3, n, B0, B1, B2, B3, B4, B5, B6, B7, B8);
            }
        }
    }
    const int koff = (lane < 16) ? 0 : 2;   // A-matrix K offset for this lane half

    const float start = t_in + 0.5f * kVL;
    const float end   = t_in + 0.5f * kVL * (float)ns;
    const float dw    = 1.0f / (float)(ns - 1);

    float* fwave = s_feat + wave * (32 * kFP);

    // ---------------- Phase A: gather + trilerp feat[28], WMMA SH contraction --------
    // Batch loop is wave-uniform (trip count depends only on wave id) so every
    // WMMA executes with full EXEC as required.
    for (int base = wave * 32; base < ns; base += kNT) {
        const int s  = base + lane;
        const int sc = (s < ns) ? s : (ns - 1);   // clamp for OOB lanes (results unstored)

        const float w = (float)sc * dw;
        float t = start * (1.0f - w) + end * w;
        t = fminf(t, t_out);
        const float px = ox + t * dx, py = oy + t * dy, pz = oz + t * dz;

        // per-axis lower/upper neighbor voxel coords, exactly as reference
        const float nlx = floorf((px - kHVL) / kVL + 1e-5f);
        const float nhx = floorf((px + kHVL) / kVL + 1e-5f);
        const float nly = floorf((py - kHVL) / kVL + 1e-5f);
        const float nhy = floorf((py + kHVL) / kVL + 1e-5f);
        const float nlz = floorf((pz - kHVL) / kVL + 1e-5f);
        const float nhz = floorf((pz + kHVL) / kVL + 1e-5f);

        const int ilx = min(max((int)(nlx + 96.0f), 0), kRes - 1);
        const int ihx = min(max((int)(nhx + 96.0f), 0), kRes - 1);
        const int ily = min(max((int)(nly + 96.0f), 0), kRes - 1);
        const int ihy = min(max((int)(nhy + 96.0f), 0), kRes - 1);
        const int ilz = min(max((int)(nlz + 96.0f), 0), kRes - 1);
        const int ihz = min(max((int)(nhz + 96.0f), 0), kRes - 1);

        // corner-0 (clipped) center -> local trilinear coordinates
        const float cx = fminf(fmaxf((nlx + 0.5f) * kVL, -1.0f + kHVL), 1.0f - kHVL);
        const float cy = fminf(fmaxf((nly + 0.5f) * kVL, -1.0f + kHVL), 1.0f - kHVL);
        const float cz = fminf(fmaxf((nlz + 0.5f) * kVL, -1.0f + kHVL), 1.0f - kHVL);
        const float xr = (px - cx) / kVL;
        const float yr = (py - cy) / kVL;
        const float zr = (pz - cz) / kVL;
        const float fx1 = xr, fx0 = 1.0f - xr;
        const float fy1 = yr, fy0 = 1.0f - yr;
        const float fz1 = zr, fz0 = 1.0f - zr;

        float f[28];
#pragma unroll
        for (int c = 0; c < 28; ++c) f[c] = 0.0f;

#pragma unroll
        for (int e = 0; e < 8; ++e) {
            const int ix = (e & 4) ? ihx : ilx;
            const int iy = (e & 2) ? ihy : ily;
            const int iz = (e & 1) ? ihz : ilz;
            const float we = ((e & 4) ? fx1 : fx0) * ((e & 2) ? fy1 : fy0) * ((e & 1) ? fz1 : fz0);
            const int flat = (ix * kRes + iy) * kRes + iz;
            const int nid  = grid_idx[flat];
            const float4* q = reinterpret_cast<const float4*>(grid_data + (size_t)nid * 28);
#pragma unroll
            for (int c = 0; c < 7; ++c) {
                const float4 v = q[c];
                f[4 * c + 0] = fmaf(we, v.x, f[4 * c + 0]);
                f[4 * c + 1] = fmaf(we, v.y, f[4 * c + 1]);
                f[4 * c + 2] = fmaf(we, v.z, f[4 * c + 2]);
                f[4 * c + 3] = fmaf(we, v.w, f[4 * c + 3]);
            }
        }

        // stage feat into this wave's LDS transpose buffer (stride 29: bank-conflict free)
        {
            float* fb = fwave + lane * kFP;
#pragma unroll
            for (int c = 0; c < 28; ++c) fb[c] = f[c];
        }
        if (s < ns) s_intr[s] = t;

        // ---- SH contraction: two 16-sample tiles, 7 chained f32 16x16x4 WMMAs each ----
#pragma unroll
        for (int half = 0; half < 2; ++half) {
            const int srow = (lane & 15) + 16 * half;      // A row M -> sample within batch
            const float* fp = fwave + srow * kFP;
            v8f acc = {};
#pragma unroll
            for (int j = 0; j < 7; ++j) {
                v2f a, bb;
                a.x  = fp[4 * j + koff];
                a.y  = fp[4 * j + koff + 1];
                bb.x = bw0[j];
                bb.y = bw1[j];
                acc = __builtin_amdgcn_wmma_f32_16x16x4_f32(
                        /*neg_a=*/false, a, /*neg_b=*/false, bb,
                        /*c_mod=*/(short)0, acc, /*reuse_a=*/false, /*reuse_b=*/false);
            }
            // D layout: lane holds column N=lane&15 for rows M=(lane<16?0..7:8..15).
            const int col = lane & 15;
            if (col < 4) {
                const int mbase = (lane >> 4) * 8;
#pragma unroll
                for (int r = 0; r < 8; ++r) {
                    const int ss = base + 16 * half + mbase + r;
                    if (ss < ns) {
                        const float v = acc[r];
                        if (col == 3) s_sig[ss] = v;
                        else          s_rgb[3 * ss + col] = v;
                    }
                }
            }
        }

        // Prefetch this lane's next sample record (hint only; gfx1250 global_prefetch_b8).
        const int s2 = s + kNT;
        if (s2 < ns) {
            const float w2 = (float)s2 * dw;
            const float t2 = fminf(start * (1.0f - w2) + end * w2, t_out);
            const float qx = ox + t2 * dx, qy = oy + t2 * dy, qz = oz + t2 * dz;
            const int jx = min(max((int)(floorf((qx - kHVL) / kVL + 1e-5f) + 96.0f), 0), kRes - 1);
            const int jy = min(max((int)(floorf((qy - kHVL) / kVL + 1e-5f) + 96.0f), 0), kRes - 1);
            const int jz = min(max((int)(floorf((qz - kHVL) / kVL + 1e-5f) + 96.0f), 0), kRes - 1);
            const size_t f2 = ((size_t)jx * kRes + jy) * kRes + jz;
            __builtin_prefetch(grid_idx + f2, 0, 3);
            const float* g2 = grid_data + f2 * 28;      // identity-map guess, hint only
            __builtin_prefetch(g2, 0, 3);
            __builtin_prefetch(g2 + 16, 0, 3);          // cover the 112B record's 2nd line
        }
    }

    __syncthreads();

    float* out_rgb   = out;
    float* out_alpha = out + (size_t)B * 3;
    float* out_depth = out + (size_t)B * 3 + (size_t)B * (ns - 1);

    // ---------------- Phase B: alpha + sigmoid(rgb), parallel over intervals ----------
    for (int i = tid; i < ns - 1; i += kNT) {
        const float dist = (s_intr[i + 1] - s_intr[i]) * dn;
        const float a = 1.0f - expf(-fmaxf(s_sig[i], 0.0f) * dist);
        s_alpha[i] = a;
        out_alpha[(size_t)b * (ns - 1) + i] = a;
        s_rgb[3 * i + 0] = 1.0f / (1.0f + expf(-s_rgb[3 * i + 0]));
        s_rgb[3 * i + 1] = 1.0f / (1.0f + expf(-s_rgb[3 * i + 1]));
        s_rgb[3 * i + 2] = 1.0f / (1.0f + expf(-s_rgb[3 * i + 2]));
    }

    __syncthreads();

    // ---------------- Phase C: sequential transmittance scan (thread 0) ---------------
    if (tid == 0) {
        float T = 1.0f, acc = 0.0f, r = 0.0f, g = 0.0f, bl = 0.0f, dep = 0.0f;
        for (int i = 0; i < ns - 1; ++i) {
            const float a  = s_alpha[i];
            const float ab = a * T;
            acc += ab;
            r   = fmaf(ab, s_rgb[3 * i + 0], r);
            g   = fmaf(ab, s_rgb[3 * i + 1], g);
            bl  = fmaf(ab, s_rgb[3 * i + 2], bl);
            dep = fmaf(ab, s_intr[i], dep);
            T *= (1.0f - a + 1e-10f);
        }
        const float bg = 1.0f - acc;     // white background
        out_rgb[3 * b + 0] = r  + bg;
        out_rgb[3 * b + 1] = g  + bg;
        out_rgb[3 * b + 2] = bl + bg;
        out_depth[b] = dep;
    }
}

extern "C" void kernel_launch(void* const* d_in, const int* in_sizes, int n_in,
                              void* d_out, int out_size, void* d_ws, size_t ws_size,
                              hipStream_t stream) {
    const float* rays_o    = (const float*)d_in[0];
    const float* rays_d    = (const float*)d_in[1];
    const float* grid_data = (const float*)d_in[2];
    const int*   grid_idx  = (const int*)d_in[3];
    // d_in[4] = n_samples (device scalar); derive ns host-side from out_size instead:
    // out_size = B*3 + B*(ns-1) + B  =>  ns = out_size/B - 3
    const int B  = in_sizes[0] / 3;
    const int ns = out_size / B - 3;
    const size_t smem = ((size_t)6 * ns + (size_t)4 * 32 * kFP) * sizeof(float);
    plenoxel_render_kernel<<<dim3(B), dim3(kNT), smem, stream>>>(
        rays_o, rays_d, grid_data, grid_idx, (float*)d_out, B, ns);
}